// Jpeg_10385230921938
// MI455X (gfx1250) — compile-verified
//
#include <hip/hip_runtime.h>
#include <math.h>

// ---------------------------------------------------------------------------
// JPEG round-trip (RGB->YCC -> 8x8 DCT -> diff-quant -> IDCT -> YCC->RGB)
// Fused single-pass kernel for gfx1250 using V_WMMA_F32_16X16X4_F32.
// Each wave32 processes one 16x16 spatial tile (= 2x2 JPEG 8x8 blocks) of one
// image, for all 3 channels. The 8x8 DCT operator is embedded block-diagonally
// in a 16x16 matrix so one 16x16 matmul performs four 8x8 transforms.
// Quantization tables are quality-folded at compile time (quality=10 ->
// qv = min(5*Q, 255) exactly), with constant reciprocals so no fdiv is needed.
// ---------------------------------------------------------------------------

typedef __attribute__((ext_vector_type(2))) float v2f;
typedef __attribute__((ext_vector_type(8))) float v8f;

// Quality-folded luma quant values: min(5*Q, 255)
__constant__ float QVL[64] = {
    80,  55,  50,  80,  120, 200, 255, 255,
    60,  60,  70,  95,  130, 255, 255, 255,
    70,  65,  80,  120, 200, 255, 255, 255,
    70,  85,  110, 145, 255, 255, 255, 255,
    90,  110, 185, 255, 255, 255, 255, 255,
    120, 175, 255, 255, 255, 255, 255, 255,
    245, 255, 255, 255, 255, 255, 255, 255,
    255, 255, 255, 255, 255, 255, 255, 255};

__constant__ float QRL[64] = {
    1.0f/80.f,  1.0f/55.f,  1.0f/50.f,  1.0f/80.f,  1.0f/120.f, 1.0f/200.f, 1.0f/255.f, 1.0f/255.f,
    1.0f/60.f,  1.0f/60.f,  1.0f/70.f,  1.0f/95.f,  1.0f/130.f, 1.0f/255.f, 1.0f/255.f, 1.0f/255.f,
    1.0f/70.f,  1.0f/65.f,  1.0f/80.f,  1.0f/120.f, 1.0f/200.f, 1.0f/255.f, 1.0f/255.f, 1.0f/255.f,
    1.0f/70.f,  1.0f/85.f,  1.0f/110.f, 1.0f/145.f, 1.0f/255.f, 1.0f/255.f, 1.0f/255.f, 1.0f/255.f,
    1.0f/90.f,  1.0f/110.f, 1.0f/185.f, 1.0f/255.f, 1.0f/255.f, 1.0f/255.f, 1.0f/255.f, 1.0f/255.f,
    1.0f/120.f, 1.0f/175.f, 1.0f/255.f, 1.0f/255.f, 1.0f/255.f, 1.0f/255.f, 1.0f/255.f, 1.0f/255.f,
    1.0f/245.f, 1.0f/255.f, 1.0f/255.f, 1.0f/255.f, 1.0f/255.f, 1.0f/255.f, 1.0f/255.f, 1.0f/255.f,
    1.0f/255.f, 1.0f/255.f, 1.0f/255.f, 1.0f/255.f, 1.0f/255.f, 1.0f/255.f, 1.0f/255.f, 1.0f/255.f};

// Quality-folded chroma quant values: min(5*Q, 255)
__constant__ float QVC[64] = {
    85,  90,  120, 235, 255, 255, 255, 255,
    90,  105, 130, 255, 255, 255, 255, 255,
    120, 130, 255, 255, 255, 255, 255, 255,
    235, 255, 255, 255, 255, 255, 255, 255,
    255, 255, 255, 255, 255, 255, 255, 255,
    255, 255, 255, 255, 255, 255, 255, 255,
    255, 255, 255, 255, 255, 255, 255, 255,
    255, 255, 255, 255, 255, 255, 255, 255};

__constant__ float QRC[64] = {
    1.0f/85.f,  1.0f/90.f,  1.0f/120.f, 1.0f/235.f, 1.0f/255.f, 1.0f/255.f, 1.0f/255.f, 1.0f/255.f,
    1.0f/90.f,  1.0f/105.f, 1.0f/130.f, 1.0f/255.f, 1.0f/255.f, 1.0f/255.f, 1.0f/255.f, 1.0f/255.f,
    1.0f/120.f, 1.0f/130.f, 1.0f/255.f, 1.0f/255.f, 1.0f/255.f, 1.0f/255.f, 1.0f/255.f, 1.0f/255.f,
    1.0f/235.f, 1.0f/255.f, 1.0f/255.f, 1.0f/255.f, 1.0f/255.f, 1.0f/255.f, 1.0f/255.f, 1.0f/255.f,
    1.0f/255.f, 1.0f/255.f, 1.0f/255.f, 1.0f/255.f, 1.0f/255.f, 1.0f/255.f, 1.0f/255.f, 1.0f/255.f,
    1.0f/255.f, 1.0f/255.f, 1.0f/255.f, 1.0f/255.f, 1.0f/255.f, 1.0f/255.f, 1.0f/255.f, 1.0f/255.f,
    1.0f/255.f, 1.0f/255.f, 1.0f/255.f, 1.0f/255.f, 1.0f/255.f, 1.0f/255.f, 1.0f/255.f, 1.0f/255.f,
    1.0f/255.f, 1.0f/255.f, 1.0f/255.f, 1.0f/255.f, 1.0f/255.f, 1.0f/255.f, 1.0f/255.f, 1.0f/255.f};

__device__ __forceinline__ v8f wmma4(v2f a, v2f b, v8f c) {
  // D = A(16x4,f32) * B(4x16,f32) + C(16x16,f32)
  return __builtin_amdgcn_wmma_f32_16x16x4_f32(
      /*neg_a=*/false, a, /*neg_b=*/false, b,
      /*c_mod=*/(short)0, c, /*reuse_a=*/false, /*reuse_b=*/false);
}

// Per-wave LDS slice is private to the wave; same-wave DS ops are processed
// in order by hardware, so we only need a compiler scheduling barrier.
__device__ __forceinline__ void wave_lds_fence() {
  __builtin_amdgcn_wave_barrier();
  asm volatile("" ::: "memory");
}

// block-diagonal 16x16 DCT operator: H16 = diag(h8, h8),
// h8(j,i) = cos((2j+1) * i * pi/16)
__device__ __forceinline__ float h16_elem(int r, int c) {
  const float PI16 = 0.19634954084936207f; // pi/16
  if ((r >> 3) != (c >> 3)) return 0.0f;
  return __cosf((float)((2 * (r & 7) + 1) * (c & 7)) * PI16);
}

__global__ __launch_bounds__(256) void jpeg_roundtrip_kernel(
    const float* __restrict__ x, float* __restrict__ out, int nImages) {
  const int tid  = threadIdx.x;
  const int lane = tid & 31;
  const int wave = tid >> 5;
  const int half = lane >> 4;  // 0: lanes 0-15, 1: lanes 16-31
  const int lc   = lane & 15;  // column / N index

  __shared__ float ldsbuf[8][16 * 17];  // 16x16 tile, padded row stride 17
  float* buf = ldsbuf[wave];

  const int tile       = blockIdx.x * 8 + wave;
  const int totalTiles = nImages * 32 * 32;
  if (tile >= totalTiles) return;  // whole wave exits together (EXEC all-1s)

  const int tx   = tile & 31;
  const int ty   = (tile >> 5) & 31;
  const int img  = tile >> 10;
  const int row0 = ty * 16;
  const int col0 = tx * 16;
  const int W = 512;
  const long planeStride = 512L * 512L;

  // --- constant DCT operator fragments (A-operand layout: M=lc, K split) ---
  v2f hT[4];  // A = H16^T : elem(m,k) = H16(k,m)
  v2f hN[4];  // A = H16   : elem(m,k) = H16(m,k)
#pragma unroll
  for (int j = 0; j < 4; ++j) {
    const int k0 = 4 * j + 2 * half;
    v2f t0 = {h16_elem(k0, lc), h16_elem(k0 + 1, lc)};
    v2f t1 = {h16_elem(lc, k0), h16_elem(lc, k0 + 1)};
    hT[j] = t0;
    hN[j] = t1;
  }

  // --- coalesced nontemporal load of the 16x16 tile, all 3 channels -------
  float rgbv[3][8];
#pragma unroll
  for (int c = 0; c < 3; ++c) {
    const float* src = x + ((long)img * 3 + c) * planeStride;
#pragma unroll
    for (int i = 0; i < 8; ++i) {
      const int rr = row0 + 2 * i + half;
      rgbv[c][i] =
          __builtin_nontemporal_load(&src[(long)rr * W + col0 + lc]) * 255.0f;
    }
  }

  float outv[3][8];
#pragma unroll
  for (int d = 0; d < 3; ++d)
#pragma unroll
    for (int r = 0; r < 8; ++r) outv[d][r] = 0.0f;

  const float invSqrt2 = 0.70710678118654752f;
  const v8f vzero = {0.f, 0.f, 0.f, 0.f, 0.f, 0.f, 0.f, 0.f};

#pragma unroll
  for (int ch = 0; ch < 3; ++ch) {
    // RGB -> centered YCC (Y-128, Cb, Cr); write tile matrix M to LDS
    const float k0c = (ch == 0) ? 0.299f : (ch == 1) ? -0.168736f : 0.5f;
    const float k1c = (ch == 0) ? 0.587f : (ch == 1) ? -0.331264f : -0.418688f;
    const float k2c = (ch == 0) ? 0.114f : (ch == 1) ? 0.5f : -0.081312f;
    const float cbias = (ch == 0) ? -128.0f : 0.0f;
#pragma unroll
    for (int i = 0; i < 8; ++i) {
      const float v =
          k0c * rgbv[0][i] + k1c * rgbv[1][i] + k2c * rgbv[2][i] + cbias;
      buf[(2 * i + half) * 17 + lc] = v;
    }
    wave_lds_fence();

    v2f bf[4];
    v8f acc;

    // ---- Stage 1: U = H16^T * M -----------------------------------------
#pragma unroll
    for (int j = 0; j < 4; ++j) {
      const int k = 4 * j + 2 * half;
      v2f t = {buf[k * 17 + lc], buf[(k + 1) * 17 + lc]};
      bf[j] = t;
    }
    acc = vzero;
#pragma unroll
    for (int j = 0; j < 4; ++j) acc = wmma4(hT[j], bf[j], acc);
    wave_lds_fence();
#pragma unroll
    for (int r = 0; r < 8; ++r) buf[(r + 8 * half) * 17 + lc] = acc[r];
    wave_lds_fence();

    // ---- Stage 2: V = H16^T * U^T  (=> V = T^T, T = H^T M H) ------------
#pragma unroll
    for (int j = 0; j < 4; ++j) {
      const int k = 4 * j + 2 * half;
      v2f t = {buf[lc * 17 + k], buf[lc * 17 + k + 1]};
      bf[j] = t;
    }
    acc = vzero;
#pragma unroll
    for (int j = 0; j < 4; ++j) acc = wmma4(hT[j], bf[j], acc);
    wave_lds_fence();

    // ---- Pointwise: quantize, diff-round, dequantize ---------------------
    // Register (r, lane) holds T(n, m): n = lc, m = r + 8*half.
    const float* qv_t = (ch == 0) ? QVL : QVC;
    const float* qr_t = (ch == 0) ? QRL : QRC;
    const int n8 = lc & 7;
    const float nrm_n = (n8 == 0) ? invSqrt2 : 1.0f;
#pragma unroll
    for (int r = 0; r < 8; ++r) {
      const float qv = qv_t[n8 * 8 + r];
      const float qr = qr_t[n8 * 8 + r];
      const float nrm = nrm_n * ((r == 0) ? invSqrt2 : 1.0f);
      const float xq = (0.25f * nrm * acc[r]) * qr;    // dctc / q
      const float rr = rintf(xq);                      // round half-to-even
      const float d  = xq - rr;
      const float deq = (rr + d * d * d) * qv;         // diff_round * q
      buf[(r + 8 * half) * 17 + lc] = nrm * deq;       // store W^T
    }
    wave_lds_fence();

    // ---- Stage 3: P = H16 * W^T ------------------------------------------
#pragma unroll
    for (int j = 0; j < 4; ++j) {
      const int k = 4 * j + 2 * half;
      v2f t = {buf[k * 17 + lc], buf[(k + 1) * 17 + lc]};
      bf[j] = t;
    }
    acc = vzero;
#pragma unroll
    for (int j = 0; j < 4; ++j) acc = wmma4(hN[j], bf[j], acc);
    wave_lds_fence();
#pragma unroll
    for (int r = 0; r < 8; ++r) buf[(r + 8 * half) * 17 + lc] = acc[r];
    wave_lds_fence();

    // ---- Stage 4: S = H16 * P^T  (= H W H^T, the spatial block) ----------
#pragma unroll
    for (int j = 0; j < 4; ++j) {
      const int k = 4 * j + 2 * half;
      v2f t = {buf[lc * 17 + k], buf[lc * 17 + k + 1]};
      bf[j] = t;
    }
    acc = vzero;
#pragma unroll
    for (int j = 0; j < 4; ++j) acc = wmma4(hN[j], bf[j], acc);
    wave_lds_fence();

    // ---- Accumulate YCC->RGB (shifted: Y keeps +128, chroma cancels) -----
    const float oshift = (ch == 0) ? 128.0f : 0.0f;
    const float m0 = (ch == 0) ? 1.0f : (ch == 1) ? 0.0f : 1.402f;
    const float m1 = (ch == 0) ? 1.0f : (ch == 1) ? -0.344136f : -0.714136f;
    const float m2 = (ch == 0) ? 1.0f : (ch == 1) ? 1.772f : 0.0f;
#pragma unroll
    for (int r = 0; r < 8; ++r) {
      const float s = 0.25f * acc[r] + oshift;
      outv[0][r] += m0 * s;
      outv[1][r] += m1 * s;
      outv[2][r] += m2 * s;
    }
  }

  // --- coalesced nontemporal store: (r, lane) -> (row0+r+8*half, col0+lc) --
  const float inv255 = 1.0f / 255.0f;
#pragma unroll
  for (int d = 0; d < 3; ++d) {
    float* dst = out + ((long)img * 3 + d) * planeStride;
#pragma unroll
    for (int r = 0; r < 8; ++r) {
      __builtin_nontemporal_store(
          outv[d][r] * inv255 - 0.5f,
          &dst[(long)(row0 + r + 8 * half) * W + col0 + lc]);
    }
  }
}

extern "C" void kernel_launch(void* const* d_in, const int* in_sizes, int n_in,
                              void* d_out, int out_size, void* d_ws,
                              size_t ws_size, hipStream_t stream) {
  const float* x = (const float*)d_in[0];
  float* out = (float*)d_out;

  const int total   = in_sizes[0];               // B*3*512*512
  const int nImages = total / (3 * 512 * 512);   // 32
  const int tiles   = nImages * 32 * 32;         // 16x16 tiles
  const int blocks  = (tiles + 7) / 8;           // 8 waves per block

  jpeg_roundtrip_kernel<<<blocks, 256, 0, stream>>>(x, out, nImages);
}